// MultiQueryRouter_25374666785274
// MI455X (gfx1250) — compile-verified
//
#include <hip/hip_runtime.h>
#include <hip/hip_bf16.h>

typedef __attribute__((ext_vector_type(2))) float v2f;
typedef __attribute__((ext_vector_type(4))) float v4f;
typedef __attribute__((ext_vector_type(8))) float v8f;

#define BATCH 4
#define NTOK  16384
#define DDIM  1024
#define RANK_ 16
#define NUMQ  8
#define KSEL  1638

// ---------------- Kernel 1: Wc[q][d] = sum_r queries[q][r] * Wk[r][d], padded to 16 rows
__global__ void combine_wq(const float* __restrict__ Wk, const float* __restrict__ q,
                           float* __restrict__ Wc) {
    int i  = blockIdx.x * blockDim.x + threadIdx.x;   // 0 .. 16*1024
    int qi = i >> 10;
    int d  = i & (DDIM - 1);
    float acc = 0.f;
    if (qi < NUMQ) {
#pragma unroll
        for (int r = 0; r < RANK_; ++r)
            acc = fmaf(q[qi * RANK_ + r], Wk[r * DDIM + d], acc);
    }
    Wc[i] = acc;
}

// ---------------- Kernel 2: token_scores[b,n] = max_q x[b,n,:] . Wc[q,:]  via f32 WMMA
__global__ void __launch_bounds__(256)
score_wmma(const float* __restrict__ x, const float* __restrict__ Wc,
           float* __restrict__ scores) {
    __shared__ float sW[16 * DDIM];              // 64 KB: 16 query rows (8 real + 8 zero)
    for (int i = threadIdx.x; i < (16 * DDIM) / 4; i += 256)
        ((v4f*)sW)[i] = ((const v4f*)Wc)[i];
    __syncthreads();

    const int lane = threadIdx.x & 31;
    const int wv   = threadIdx.x >> 5;
    const int tile = blockIdx.x * 8 + wv;        // 16 tokens per tile, 4096 tiles
    const int m    = lane & 15;                  // A: row M;  B: col N (query)
    const int half = lane >> 4;
    const int kh   = half * 2;                   // K sub-offset within chunk (A & B layouts)

    const float* ap = x  + (size_t)tile * (16 * DDIM) + (size_t)m * DDIM + kh;
    const float* bp = sW + m * DDIM + kh;

    v8f c0 = {}, c1 = {}, c2 = {}, c3 = {};      // 4 accumulators to break WMMA RAW chain
#pragma unroll 4
    for (int kb = 0; kb < DDIM; kb += 16) {
        v2f a0 = *(const v2f*)(ap + kb + 0);
        v2f b0 = *(const v2f*)(bp + kb + 0);
        c0 = __builtin_amdgcn_wmma_f32_16x16x4_f32(false, a0, false, b0, (short)0, c0, false, false);
        v2f a1 = *(const v2f*)(ap + kb + 4);
        v2f b1 = *(const v2f*)(bp + kb + 4);
        c1 = __builtin_amdgcn_wmma_f32_16x16x4_f32(false, a1, false, b1, (short)0, c1, false, false);
        v2f a2 = *(const v2f*)(ap + kb + 8);
        v2f b2 = *(const v2f*)(bp + kb + 8);
        c2 = __builtin_amdgcn_wmma_f32_16x16x4_f32(false, a2, false, b2, (short)0, c2, false, false);
        v2f a3 = *(const v2f*)(ap + kb + 12);
        v2f b3 = *(const v2f*)(bp + kb + 12);
        c3 = __builtin_amdgcn_wmma_f32_16x16x4_f32(false, a3, false, b3, (short)0, c3, false, false);
    }

    // C layout: VGPR r, lanes0-15 -> M=r, N=lane; lanes16-31 -> M=r+8, N=lane-16.
    // Max over q = N in [0,8): xor-reduce over lane bits 0..2.
#pragma unroll
    for (int r = 0; r < 8; ++r) {
        float v = c0[r] + c1[r] + c2[r] + c3[r];
        v = fmaxf(v, __shfl_xor(v, 1));
        v = fmaxf(v, __shfl_xor(v, 2));
        v = fmaxf(v, __shfl_xor(v, 4));
        if ((lane & 15) == 0)
            scores[tile * 16 + r + 8 * half] = v;
    }
}

// ---------------- Kernel 3: per-batch exact top-K (radix select + ordered emit)
__global__ void __launch_bounds__(1024)
topk_select(const float* __restrict__ scores, int* __restrict__ out) {
    const int b = blockIdx.x;
    const int t = threadIdx.x;
    const float* s = scores + b * NTOK;

    __shared__ unsigned hist[256];
    __shared__ unsigned scn[1024];
    __shared__ unsigned s_prefix, s_R;

    unsigned key[16];                            // order-preserving u32 encoding
#pragma unroll
    for (int j = 0; j < 16; ++j) {
        unsigned u = __float_as_uint(s[t * 16 + j]);
        key[j] = u ^ ((u >> 31) ? 0xFFFFFFFFu : 0x80000000u);
    }

    if (t == 0) { s_prefix = 0u; s_R = KSEL; }

    for (int p = 3; p >= 0; --p) {
        unsigned maskhi = (p == 3) ? 0u : (0xFFFFFFFFu << ((p + 1) * 8));
        if (t < 256) hist[t] = 0u;
        __syncthreads();
        unsigned prefix = s_prefix;
#pragma unroll
        for (int j = 0; j < 16; ++j)
            if (((key[j] ^ prefix) & maskhi) == 0u)
                atomicAdd(&hist[(key[j] >> (p * 8)) & 0xFFu], 1u);
        __syncthreads();
        if (t == 0) {
            unsigned R = s_R, cum = 0u; int sel = 0;
            for (int bin = 255; bin >= 0; --bin) {
                unsigned c = hist[bin];
                if (cum + c >= R) { sel = bin; s_R = R - cum; break; }
                cum += c;
            }
            s_prefix = prefix | ((unsigned)sel << (p * 8));
        }
        __syncthreads();
    }
    const unsigned thresh = s_prefix;            // K-th largest key
    const unsigned R      = s_R;                 // ties to take (earliest indices first)

    unsigned acnt = 0, tcnt = 0;
#pragma unroll
    for (int j = 0; j < 16; ++j) {
        acnt += (key[j] > thresh);
        tcnt += (key[j] == thresh);
    }
    unsigned packed = (acnt << 16) | tcnt;       // fields <= 16384, no carry between them
    scn[t] = packed;
    __syncthreads();
    for (int off = 1; off < 1024; off <<= 1) {   // Hillis-Steele inclusive scan
        unsigned v   = scn[t];
        unsigned add = (t >= off) ? scn[t - off] : 0u;
        __syncthreads();
        scn[t] = v + add;
        __syncthreads();
    }
    unsigned excl = scn[t] - packed;
    unsigned A = excl >> 16;                     // # '>' before my first element
    unsigned T = excl & 0xFFFFu;                 // # '==' before my first element

#pragma unroll
    for (int j = 0; j < 16; ++j) {
        if (key[j] > thresh) {
            out[b * KSEL + (A + (T < R ? T : R))] = t * 16 + j;
            ++A;
        } else if (key[j] == thresh) {
            if (T < R) out[b * KSEL + (A + T)] = t * 16 + j;
            ++T;
        }
    }
}

extern "C" void kernel_launch(void* const* d_in, const int* in_sizes, int n_in,
                              void* d_out, int out_size, void* d_ws, size_t ws_size,
                              hipStream_t stream) {
    const float* x  = (const float*)d_in[0];   // [B, N, D]
    const float* Wk = (const float*)d_in[1];   // [RANK, D]
    const float* qs = (const float*)d_in[2];   // [NUM_Q, RANK]
    int* out = (int*)d_out;                    // [B, KSEL]

    float* Wc     = (float*)d_ws;                                   // 16*1024 f32 (64 KB)
    float* scores = (float*)((char*)d_ws + 16 * DDIM * sizeof(float)); // B*N f32 (256 KB)

    combine_wq<<<(16 * DDIM) / 256, 256, 0, stream>>>(Wk, qs, Wc);
    score_wmma<<<(BATCH * NTOK / 16) / 8, 256, 0, stream>>>(x, Wc, scores);
    topk_select<<<BATCH, 1024, 0, stream>>>(scores, out);
}